// NGCFModel_17343077941930
// MI455X (gfx1250) — compile-verified
//
#include <hip/hip_runtime.h>
#include <hip/hip_bf16.h>

// ---------------------------------------------------------------------------
// NGCF forward on MI455X (gfx1250, wave32).
//   E  : [200000 x 64] f32 node embeddings (ping-pong in workspace)
//   R  : [200000 x 64] f32 SpMM accumulator (both folds fused: linearity)
//   per layer: R = (A0+A1) @ E      (COO scatter, hw f32 atomics)
//              E' = leaky( R@W1 + (R.*E)@W2 )   (V_WMMA_F32_16X16X4_F32)
//   out[n] += dot(E'[n], E'[nu+n]) after every layer (incl. layer 0)
//
// LDS holds W1/W2 pre-swizzled into WMMA-fragment order so each lane fetches
// both B fragments for a k-step with ONE ds_load_b128 (no repacking movs).
// ---------------------------------------------------------------------------

#define NUM_USERS 100000
#define NUM_ITEMS 100000
#define NT        (NUM_USERS + NUM_ITEMS)
#define EMB       64
#define N_LAYERS  3
#define NNZ_TOTAL 3200000   // N_FOLD * NNZ_PER_FOLD, folds stored contiguously
#define ALPHA     0.2f

typedef __attribute__((ext_vector_type(2))) float v2f;
typedef __attribute__((ext_vector_type(4))) float v4f;
typedef __attribute__((ext_vector_type(8))) float v8f;

// ---- E0 = concat(user_table[user_idx], item_table[item_idx]) --------------
__global__ void init_emb_kernel(const float* __restrict__ ut,
                                const float* __restrict__ it,
                                const int* __restrict__ ui,
                                const int* __restrict__ ii,
                                float* __restrict__ E) {
    long gid = (long)blockIdx.x * blockDim.x + threadIdx.x;   // (row, 16B chunk)
    long total = (long)NT * (EMB / 4);
    if (gid >= total) return;
    int row = (int)(gid >> 4);
    int c4  = (int)(gid & 15) * 4;
    const float* src = (row < NUM_USERS)
        ? ut + (size_t)ui[row] * EMB
        : it + (size_t)ii[row - NUM_USERS] * EMB;
    *(float4*)(E + (size_t)row * EMB + c4) = *(const float4*)(src + c4);
}

// ---- zero the SpMM accumulator --------------------------------------------
__global__ void zero_kernel(float* __restrict__ p, long n4) {
    long i = (long)blockIdx.x * blockDim.x + threadIdx.x;
    if (i < n4) *(float4*)(p + i * 4) = make_float4(0.f, 0.f, 0.f, 0.f);
}

// ---- COO SpMM scatter: one wave per edge, 2 floats per lane ---------------
__global__ void spmm_scatter_kernel(const float* __restrict__ vals,
                                    const int* __restrict__ rows,
                                    const int* __restrict__ cols,
                                    const float* __restrict__ E,
                                    float* __restrict__ R) {
    long gid  = (long)blockIdx.x * blockDim.x + threadIdx.x;
    long edge = gid >> 5;
    int  lane = (int)(gid & 31);
    if (edge >= (long)NNZ_TOTAL) return;
    int   c = cols[edge];
    int   r = rows[edge];
    float v = vals[edge];
    v2f e = *(const v2f*)(E + (size_t)c * EMB + lane * 2);
    float* dst = R + (size_t)r * EMB + lane * 2;
    // relaxed agent-scope fadd -> global_atomic_add_f32 (no return)
    __hip_atomic_fetch_add(dst + 0, e.x * v, __ATOMIC_RELAXED,
                           __HIP_MEMORY_SCOPE_AGENT);
    __hip_atomic_fetch_add(dst + 1, e.y * v, __ATOMIC_RELAXED,
                           __HIP_MEMORY_SCOPE_AGENT);
}

// ---- fused dual-GEMM + gate + leaky-ReLU via V_WMMA_F32_16X16X4_F32 -------
// One wave owns a 16-row tile: Eout = leaky( R@W1 + (R.*E)@W2 ).
// A frag (16x4 f32): lane l<16 -> rows, v0/v1 = K off+0/off+1; lanes 16..31
// carry K off+2/off+3 (off = 4k + 2*half).  B frag mirrors in K.  C/D: VGPR r
// holds rows r (lanes 0-15) and r+8 (lanes 16-31).
//
// LDS fragment layout: for K-pair p = 0..31 and column n = 0..63:
//   Ws[p*256 + n*4 + {0,1,2,3}] = { W1[2p][n], W1[2p+1][n],
//                                   W2[2p][n], W2[2p+1][n] }
// so lane (half,l16) at k-step k reads one b128 at p = 2k+half, n = n0.
__global__ void __launch_bounds__(128)
gemm_layer_kernel(const float* __restrict__ R,
                  const float* __restrict__ E,
                  const float* __restrict__ W1,
                  const float* __restrict__ W2,
                  float* __restrict__ Eout) {
    __shared__ float Ws[32 * 256];     // 32 KB: interleaved W1/W2 fragments

    // cooperative swizzled fill: idx enumerates (k,n) of the 64x64 matrices
    for (int idx = threadIdx.x; idx < EMB * EMB; idx += blockDim.x) {
        int k = idx >> 6;              // 0..63
        int n = idx & 63;
        int p = k >> 1;
        int rbit = k & 1;
        float* dst = Ws + p * 256 + n * 4;
        dst[rbit]     = W1[k * EMB + n];
        dst[2 + rbit] = W2[k * EMB + n];
    }
    __syncthreads();

    const int lane = threadIdx.x & 31;
    const int wave = threadIdx.x >> 5;
    const int half = lane >> 4;        // 0: K lo pair, 1: K hi pair
    const int l16  = lane & 15;

    const int tile = blockIdx.x * 4 + wave;     // 12500 tiles == 3125*4
    const int row0 = tile * 16;

    const float* Rrow = R + (size_t)(row0 + l16) * EMB;
    const float* Erow = E + (size_t)(row0 + l16) * EMB;

    // A fragments for all 16 k-steps: plain R and gated R.*E
    v2f ar[16], ag[16];
#pragma unroll
    for (int k = 0; k < 16; ++k) {
        const int off = k * 4 + half * 2;
        v2f r = *(const v2f*)(Rrow + off);
        v2f e = *(const v2f*)(Erow + off);
        ar[k] = r;
        ag[k] = r * e;
    }

#pragma unroll
    for (int n = 0; n < 4; ++n) {
        const int n0 = n * 16 + l16;
        const float* Wn = Ws + (size_t)n0 * 4 + (size_t)half * 256;
        v8f acc = {};
#pragma unroll
        for (int k = 0; k < 16; ++k) {
            // one b128: {b1.x, b1.y, b2.x, b2.y} for K-pair p = 2k+half
            v4f b = *(const v4f*)(Wn + (size_t)k * 512);
            v2f b1; b1.x = b.x; b1.y = b.y;
            v2f b2; b2.x = b.z; b2.y = b.w;
            acc = __builtin_amdgcn_wmma_f32_16x16x4_f32(
                false, ar[k], false, b1, (short)0, acc, false, false);
            acc = __builtin_amdgcn_wmma_f32_16x16x4_f32(
                false, ag[k], false, b2, (short)0, acc, false, false);
        }
        // leaky-ReLU + store 16x16 tile
#pragma unroll
        for (int r = 0; r < 8; ++r) {
            float v = acc[r];
            v = v > 0.f ? v : ALPHA * v;
            Eout[(size_t)(row0 + r + half * 8) * EMB + n0] = v;
        }
    }
}

// ---- out[n] (+)= dot(E[n], E[nu+n]) ---------------------------------------
__global__ void dot_kernel(const float* __restrict__ E,
                           float* __restrict__ out,
                           int accumulate) {
    int n = blockIdx.x * blockDim.x + threadIdx.x;
    if (n >= NUM_USERS) return;
    const float4* u = (const float4*)(E + (size_t)n * EMB);
    const float4* v = (const float4*)(E + (size_t)(NUM_USERS + n) * EMB);
    float s = 0.f;
#pragma unroll
    for (int j = 0; j < EMB / 4; ++j) {
        float4 a = u[j], b = v[j];
        s += a.x * b.x + a.y * b.y + a.z * b.z + a.w * b.w;
    }
    if (accumulate) out[n] += s; else out[n] = s;
}

extern "C" void kernel_launch(void* const* d_in, const int* in_sizes, int n_in,
                              void* d_out, int out_size, void* d_ws, size_t ws_size,
                              hipStream_t stream) {
    const float* user_table = (const float*)d_in[0];
    const float* item_table = (const float*)d_in[1];
    const float* W1         = (const float*)d_in[2];   // [3,64,64]
    const float* W2         = (const float*)d_in[3];   // [3,64,64]
    const float* adj_vals   = (const float*)d_in[4];   // [2,NNZ] contiguous
    const int*   adj_rows   = (const int*)d_in[5];
    const int*   adj_cols   = (const int*)d_in[6];
    const int*   user_idx   = (const int*)d_in[7];
    const int*   item_idx   = (const int*)d_in[8];
    float* out = (float*)d_out;

    const size_t NE = (size_t)NT * EMB;                // 12.8M floats
    float* E0 = (float*)d_ws;
    float* E1 = E0 + NE;
    float* R  = E1 + NE;

    // layer-0 embeddings + their dot contribution
    {
        long total = (long)NT * (EMB / 4);
        int blocks = (int)((total + 255) / 256);
        init_emb_kernel<<<blocks, 256, 0, stream>>>(user_table, item_table,
                                                    user_idx, item_idx, E0);
        dot_kernel<<<(NUM_USERS + 255) / 256, 256, 0, stream>>>(E0, out, 0);
    }

    float* Ecur = E0;
    float* Enext = E1;
    for (int l = 0; l < N_LAYERS; ++l) {
        // R = 0
        long n4 = (long)NE / 4;
        zero_kernel<<<(int)((n4 + 255) / 256), 256, 0, stream>>>(R, n4);
        // R += (A0 + A1) @ Ecur  (both folds are contiguous: one launch)
        long threads = (long)NNZ_TOTAL * 32;
        spmm_scatter_kernel<<<(int)((threads + 255) / 256), 256, 0, stream>>>(
            adj_vals, adj_rows, adj_cols, Ecur, R);
        // Enext = leaky( R@W1[l] + (R.*Ecur)@W2[l] )
        gemm_layer_kernel<<<NT / 16 / 4, 128, 0, stream>>>(
            R, Ecur, W1 + (size_t)l * EMB * EMB, W2 + (size_t)l * EMB * EMB, Enext);
        // out += per-row user/item dot of this layer's embeddings
        dot_kernel<<<(NUM_USERS + 255) / 256, 256, 0, stream>>>(Enext, out, 1);
        float* t = Ecur; Ecur = Enext; Enext = t;
    }
}